// GatedLinearAttention_61495341744446
// MI455X (gfx1250) — compile-verified
//
#include <hip/hip_runtime.h>
#include <cmath>

// GLA: D_MODEL=1024, H=16, D=64, B=4, L=4096, INNER=1024
#define GLA_M     16384   // B*L
#define GLA_N     1024
#define GLA_K     1024
#define GLA_B     4
#define GLA_L     4096
#define GLA_H     16
#define GLA_D     64
#define CHUNK     64
#define NCHUNK    (GLA_L / CHUNK)

typedef float v2f __attribute__((ext_vector_type(2)));
typedef float v8f __attribute__((ext_vector_type(8)));
typedef unsigned int v4u __attribute__((ext_vector_type(4)));
typedef int v4i __attribute__((ext_vector_type(4)));
typedef int v8i __attribute__((ext_vector_type(8)));

// ---------------------------------------------------------------------------
// Dense GEMM: C = X(MxK) @ W(KxN), fp32 WMMA. mode 1 fuses sigmoid(C + bias).
// ---------------------------------------------------------------------------
#define MT 64
#define NT 128
#define KT 16

__global__ __launch_bounds__(256) void gla_gemm_f32(
    const float* __restrict__ X, const float* __restrict__ W,
    const float* __restrict__ bias, float* __restrict__ Y,
    int M, int N, int K, int mode)
{
    __shared__ float lx[MT][KT + 4];
    __shared__ float lw[KT][NT + 4];

    const int tid  = threadIdx.x;
    const int lane = tid & 31;
    const int wave = tid >> 5;
    const int wm   = wave >> 2;
    const int wn   = wave & 3;
    const int bm   = blockIdx.x * MT;
    const int bn   = blockIdx.y * NT;

    v8f acc[2][2];
    #pragma unroll
    for (int i = 0; i < 2; ++i)
        #pragma unroll
        for (int j = 0; j < 2; ++j)
            acc[i][j] = (v8f)(0.0f);

    const int xr = tid >> 2, xc = (tid & 3) * 4;
    const int wr = tid >> 5, wc = (tid & 31) * 4;

    for (int k0 = 0; k0 < K; k0 += KT) {
        *(float4*)&lx[xr][xc]    = *(const float4*)(X + (size_t)(bm + xr) * K + k0 + xc);
        *(float4*)&lw[wr][wc]    = *(const float4*)(W + (size_t)(k0 + wr)     * N + bn + wc);
        *(float4*)&lw[wr + 8][wc]= *(const float4*)(W + (size_t)(k0 + wr + 8) * N + bn + wc);
        if (k0 + KT < K) {   // prefetch next K tile -> global_prefetch_b8
            __builtin_prefetch(X + (size_t)(bm + xr) * K + k0 + KT + xc, 0, 1);
            __builtin_prefetch(W + (size_t)(k0 + KT + wr) * N + bn + wc, 0, 1);
        }
        __syncthreads();

        #pragma unroll
        for (int kk = 0; kk < KT; kk += 4) {
            const int ka = kk + ((lane >> 4) << 1);
            #pragma unroll
            for (int i = 0; i < 2; ++i) {
                const int m = wm * 32 + i * 16 + (lane & 15);
                v2f a; a.x = lx[m][ka]; a.y = lx[m][ka + 1];
                #pragma unroll
                for (int j = 0; j < 2; ++j) {
                    const int n = wn * 32 + j * 16 + (lane & 15);
                    v2f b; b.x = lw[ka][n]; b.y = lw[ka + 1][n];
                    acc[i][j] = __builtin_amdgcn_wmma_f32_16x16x4_f32(
                        false, a, false, b, (short)0, acc[i][j], false, false);
                }
            }
        }
        __syncthreads();
    }

    const int mrow_hi = (lane >> 4) << 3;
    #pragma unroll
    for (int i = 0; i < 2; ++i) {
        #pragma unroll
        for (int j = 0; j < 2; ++j) {
            const int col = bn + wn * 32 + j * 16 + (lane & 15);
            #pragma unroll
            for (int r = 0; r < 8; ++r) {
                const int row = bm + wm * 32 + i * 16 + r + mrow_hi;
                float v = acc[i][j][r];
                if (mode == 1) {
                    v = v + bias[col];
                    v = __builtin_amdgcn_rcpf(1.0f + __expf(-v));
                }
                Y[(size_t)row * N + col] = v;
            }
        }
    }
}

// ---------------------------------------------------------------------------
// TDM: load one 64x64 fp32 tile (row stride 1024 elems) into LDS with
// 4-DWORD row padding (64 -> 68 floats/row). One descriptor per call.
// ---------------------------------------------------------------------------
__device__ __forceinline__ void tdm_load_tile64(unsigned lds_addr, const float* gsrc) {
    const unsigned long long ga = (unsigned long long)(uintptr_t)gsrc;
    v4u g0;
    g0[0] = 1u;                                          // count=1, user mode
    g0[1] = lds_addr;                                    // LDS byte address
    g0[2] = (unsigned)(ga & 0xffffffffu);                // global_addr[31:0]
    g0[3] = (unsigned)((ga >> 32) & 0x01ffffffu)         // global_addr[56:32]
          | (2u << 30);                                  // type=2 (image)
    v8i g1;
    g1[0] = (2 << 16)      // data_size = 4 bytes
          | (1 << 20)      // pad_enable
          | (5 << 22)      // pad_interval: 64 DWORDs
          | (3 << 25);     // pad_amount: 4 DWORDs
    g1[1] = (CHUNK << 16);                               // tensor_dim0 = 64
    g1[2] = (CHUNK << 16);                               // tensor_dim1 = 64
    g1[3] = (CHUNK << 16);                               // tile_dim0 = 64
    g1[4] = CHUNK;                                       // tile_dim1 = 64 (tile_dim2 = 0)
    g1[5] = GLA_H * GLA_D;                               // tensor_dim0_stride = 1024 elems
    g1[6] = 0; g1[7] = 0;
    const v4i gz = {0, 0, 0, 0};
#if __clang_major__ >= 23
    const v8i gz8 = {0, 0, 0, 0, 0, 0, 0, 0};
    __builtin_amdgcn_tensor_load_to_lds(g0, g1, gz, gz, gz8, 0);
#else
    __builtin_amdgcn_tensor_load_to_lds(g0, g1, gz, gz, 0);
#endif
}

// ---------------------------------------------------------------------------
// Chunked GLA scan: one block per (b,h); 4 waves; per 64-step chunk:
//   A = Q K^T (causal), O = exp(LP) . (A_mask Vt + Q S), S = exp(LPend) . (S + K^T Vt)
// All four 64x64x64 products run on v_wmma_f32_16x16x4_f32; staging via TDM.
// ---------------------------------------------------------------------------
#define CPAD 68

__device__ __forceinline__ v2f frag_a(const float (*M)[CPAD], int m0, int kb, int lane) {
    const int m = m0 + (lane & 15);
    const int k = kb + ((lane >> 4) << 1);
    v2f a; a.x = M[m][k]; a.y = M[m][k + 1]; return a;
}
__device__ __forceinline__ v2f frag_aT(const float (*M)[CPAD], int m0, int kb, int lane) {
    const int m = m0 + (lane & 15);
    const int k = kb + ((lane >> 4) << 1);
    v2f a; a.x = M[k][m]; a.y = M[k + 1][m]; return a;
}
__device__ __forceinline__ v2f frag_b(const float (*M)[CPAD], int kb, int n0, int lane) {
    const int k = kb + ((lane >> 4) << 1);
    const int n = n0 + (lane & 15);
    v2f b; b.x = M[k][n]; b.y = M[k + 1][n]; return b;
}
__device__ __forceinline__ v2f frag_bT(const float (*M)[CPAD], int kb, int n0, int lane) {
    const int k = kb + ((lane >> 4) << 1);
    const int n = n0 + (lane & 15);
    v2f b; b.x = M[n][k]; b.y = M[n][k + 1]; return b;
}
#define WMMA4(a, b, c) __builtin_amdgcn_wmma_f32_16x16x4_f32(false, (a), false, (b), (short)0, (c), false, false)

__global__ __launch_bounds__(128) void gla_scan_chunked(
    const float* __restrict__ Qg, const float* __restrict__ Kg,
    const float* __restrict__ Vg, const float* __restrict__ Gg,
    float* __restrict__ Og)
{
    __shared__ float sQ[CHUNK][CPAD];
    __shared__ float sK[CHUNK][CPAD];
    __shared__ float sV[CHUNK][CPAD];   // becomes Vtilde in place
    __shared__ float sG[CHUNK][CPAD];
    __shared__ float sA[CHUNK][CPAD];
    __shared__ float sS[GLA_D][CPAD];
    __shared__ float sLP[CHUNK][CPAD];

    const int tid  = threadIdx.x;
    const int lane = tid & 31;
    const int w    = tid >> 5;           // 0..3, owns output rows [16w,16w+16)
    const int bh   = blockIdx.x;
    const int b    = bh >> 4;
    const int h    = bh & 15;
    const size_t headoff = (size_t)b * GLA_L * (GLA_H * GLA_D) + (size_t)h * GLA_D;

    for (int i = tid; i < GLA_D * GLA_D; i += 128) sS[i >> 6][i & 63] = 0.0f;
    __syncthreads();

    const int mrow = (lane >> 4) << 3;   // 0 or 8

    for (int c = 0; c < NCHUNK; ++c) {
        const size_t coff = headoff + (size_t)c * CHUNK * (GLA_H * GLA_D);

        // ---- stage Q, K, V, G chunk tiles via Tensor Data Mover ----
        if (w == 0) {
            tdm_load_tile64((unsigned)(uintptr_t)&sQ[0][0], Qg + coff);
            tdm_load_tile64((unsigned)(uintptr_t)&sK[0][0], Kg + coff);
            tdm_load_tile64((unsigned)(uintptr_t)&sV[0][0], Vg + coff);
            tdm_load_tile64((unsigned)(uintptr_t)&sG[0][0], Gg + coff);
            __builtin_amdgcn_s_wait_tensorcnt(0);
        }
        __syncthreads();

        // ---- gate log-cumsum per value column e = tid (threads 0..63) ----
        if (tid < GLA_D) {
            float lp = 0.0f;
            for (int t = 0; t < CHUNK; ++t) {
                lp += __logf(sG[t][tid]);
                sLP[t][tid] = lp;
            }
        }
        __syncthreads();

        // ---- V -> Vtilde = V * exp(-LP), in place ----
        for (int i = tid; i < CHUNK * GLA_D; i += 128) {
            const int t = i >> 6, e = i & 63;
            sV[t][e] = sV[t][e] * __expf(-sLP[t][e]);
        }

        // ---- GEMM1: A = Q @ K^T (rows 16w..16w+16, all 4 col tiles) ----
        v8f acc[4];
        #pragma unroll
        for (int j = 0; j < 4; ++j) acc[j] = (v8f)(0.0f);
        #pragma unroll
        for (int kb = 0; kb < GLA_D; kb += 4) {
            const v2f a = frag_a(sQ, w * 16, kb, lane);
            #pragma unroll
            for (int j = 0; j < 4; ++j)
                acc[j] = WMMA4(a, frag_bT(sK, kb, j * 16, lane), acc[j]);
        }
        __syncthreads();   // Vtilde complete before GEMM2; sA safe to write
        #pragma unroll
        for (int j = 0; j < 4; ++j)
            #pragma unroll
            for (int r = 0; r < 8; ++r) {
                const int T = w * 16 + r + mrow;
                const int t = j * 16 + (lane & 15);
                sA[T][t] = (t <= T) ? acc[j][r] : 0.0f;   // causal, diagonal inclusive
            }
        __syncthreads();

        // ---- O = exp(LP) * (A_mask @ Vt + Q @ S) ----
        #pragma unroll
        for (int j = 0; j < 4; ++j) acc[j] = (v8f)(0.0f);
        #pragma unroll
        for (int kb = 0; kb < CHUNK; kb += 4) {
            const v2f a1 = frag_a(sA, w * 16, kb, lane);
            const v2f a2 = frag_a(sQ, w * 16, kb, lane);
            #pragma unroll
            for (int j = 0; j < 4; ++j) {
                acc[j] = WMMA4(a1, frag_b(sV, kb, j * 16, lane), acc[j]);
                acc[j] = WMMA4(a2, frag_b(sS, kb, j * 16, lane), acc[j]);
            }
        }
        #pragma unroll
        for (int j = 0; j < 4; ++j)
            #pragma unroll
            for (int r = 0; r < 8; ++r) {
                const int T = w * 16 + r + mrow;
                const int e = j * 16 + (lane & 15);
                Og[coff + (size_t)T * (GLA_H * GLA_D) + e] = __expf(sLP[T][e]) * acc[j][r];
            }

        // ---- Snew = exp(LP_end) * (S + K^T @ Vt) ----
        #pragma unroll
        for (int j = 0; j < 4; ++j)
            #pragma unroll
            for (int r = 0; r < 8; ++r)
                acc[j][r] = sS[w * 16 + r + mrow][j * 16 + (lane & 15)];
        #pragma unroll
        for (int kb = 0; kb < CHUNK; kb += 4) {
            const v2f a = frag_aT(sK, w * 16, kb, lane);   // (K^T)[d][t] = sK[t][d]
            #pragma unroll
            for (int j = 0; j < 4; ++j)
                acc[j] = WMMA4(a, frag_b(sV, kb, j * 16, lane), acc[j]);
        }
        __syncthreads();   // all reads of sS done before overwrite
        #pragma unroll
        for (int j = 0; j < 4; ++j)
            #pragma unroll
            for (int r = 0; r < 8; ++r) {
                const int d = w * 16 + r + mrow;
                const int e = j * 16 + (lane & 15);
                sS[d][e] = __expf(sLP[CHUNK - 1][e]) * acc[j][r];
            }
        __syncthreads();   // S + LDS tiles stable before next chunk's TDM staging
    }
}

extern "C" void kernel_launch(void* const* d_in, const int* in_sizes, int n_in,
                              void* d_out, int out_size, void* d_ws, size_t ws_size,
                              hipStream_t stream) {
    const float* x  = (const float*)d_in[0];
    const float* Wq = (const float*)d_in[1];
    const float* Wk = (const float*)d_in[2];
    const float* Wv = (const float*)d_in[3];
    const float* Wg = (const float*)d_in[4];
    const float* bg = (const float*)d_in[5];
    const float* Wo = (const float*)d_in[6];
    float* out = (float*)d_out;

    const size_t MAT = (size_t)GLA_M * GLA_N * sizeof(float);   // 64 MB
    char* ws = (char*)d_ws;
    float* Q  = (float*)(ws + 0 * MAT);
    float* Kk = (float*)(ws + 1 * MAT);
    float* Vv = (float*)(ws + 2 * MAT);
    float* Gg = (float*)(ws + 3 * MAT);
    float* Oo = (float*)(ws + 4 * MAT);

    dim3 grid(GLA_M / MT, GLA_N / NT);   // 256 x 8
    dim3 blk(256);

    gla_gemm_f32<<<grid, blk, 0, stream>>>(x, Wq, nullptr, Q,  GLA_M, GLA_N, GLA_K, 0);
    gla_gemm_f32<<<grid, blk, 0, stream>>>(x, Wk, nullptr, Kk, GLA_M, GLA_N, GLA_K, 0);
    gla_gemm_f32<<<grid, blk, 0, stream>>>(x, Wv, nullptr, Vv, GLA_M, GLA_N, GLA_K, 0);
    gla_gemm_f32<<<grid, blk, 0, stream>>>(x, Wg, bg,      Gg, GLA_M, GLA_N, GLA_K, 1);

    gla_scan_chunked<<<dim3(GLA_B * GLA_H), dim3(128), 0, stream>>>(Q, Kk, Vv, Gg, Oo);

    gla_gemm_f32<<<grid, blk, 0, stream>>>(Oo, Wo, nullptr, out, GLA_M, GLA_N, GLA_K, 0);
}